// GraphConv_12970801234584
// MI455X (gfx1250) — compile-verified
//
#include <hip/hip_runtime.h>

// ---------------------------------------------------------------------------
// GraphConv for MI455X (gfx1250, wave32):
//   1) gemm_support_wmma : support = inp @ W      (V_WMMA_F32_16X16X4_F32)
//   2) init_out_bias     : out[r][c] = bias[c]
//   3) spmm_scatter      : out[row] += val * support[col]  (native f32 atomics,
//                          L2-resident: support+out = ~102MB < 192MB L2)
// ---------------------------------------------------------------------------

typedef __attribute__((ext_vector_type(2))) float v2f;
typedef __attribute__((ext_vector_type(8))) float v8f;

#define IN_DIM  128
#define OUT_DIM 128

// One wave per 16-row strip of the output. 8 column tiles (8*16 = 128 cols),
// K swept in steps of 4 with V_WMMA_F32_16X16X4_F32.
// VGPR striping per CDNA5 ISA 7.12.2:
//   A 16x4 f32 : lanes 0-15 -> M=lane, {K=k0,k0+1}; lanes 16-31 -> {K=k0+2,k0+3}
//   B 4x16 f32 : VGPR0 = row K (lanes 0-15) / row K+2 (lanes 16-31); VGPR1 = K+1 / K+3
//   C/D 16x16  : lane15 = N, VGPR v = M=v (lanes 0-15) / M=v+8 (lanes 16-31)
__global__ void gemm_support_wmma(const float* __restrict__ inp,
                                  const float* __restrict__ W,
                                  float* __restrict__ support,
                                  int nStrips) {
  const int lane  = threadIdx.x & 31;
  const int strip = blockIdx.x * (blockDim.x >> 5) + (threadIdx.x >> 5);
  if (strip >= nStrips) return;          // wave-uniform: EXEC stays all-ones for WMMA

  const int m0   = strip * 16;
  const int h    = lane >> 4;            // half-wave select
  const int l    = lane & 15;
  const int kofs = h * 2;

  v8f acc[8];
#pragma unroll
  for (int j = 0; j < 8; ++j) acc[j] = (v8f){0.f, 0.f, 0.f, 0.f, 0.f, 0.f, 0.f, 0.f};

  const float* arow = inp + (size_t)(m0 + l) * IN_DIM;

  for (int k0 = 0; k0 < IN_DIM; k0 += 4) {
    // A fragment: contiguous pair -> 8B-aligned b64 load (k0+kofs is even)
    const float2 av = *(const float2*)(arow + k0 + kofs);
    v2f a; a.x = av.x; a.y = av.y;
#pragma unroll
    for (int j = 0; j < 8; ++j) {
      // B fragment: each half-wave reads 16 consecutive floats of a W row
      // (64B segments; W is 64KB -> resident in WGP$/L2 after first pass)
      v2f b;
      b.x = W[(size_t)(k0 + kofs)     * OUT_DIM + j * 16 + l];
      b.y = W[(size_t)(k0 + kofs + 1) * OUT_DIM + j * 16 + l];
      // 8 args: (neg_a, A, neg_b, B, c_mod, C, reuse_a, reuse_b)
      acc[j] = __builtin_amdgcn_wmma_f32_16x16x4_f32(
          false, a, false, b, (short)0, acc[j], false, false);
    }
  }

#pragma unroll
  for (int j = 0; j < 8; ++j) {
#pragma unroll
    for (int v = 0; v < 8; ++v) {
      const int row = m0 + v + h * 8;
      support[(size_t)row * OUT_DIM + j * 16 + l] = acc[j][v];
    }
  }
}

// out[r*128 + c] = bias[c], vectorized as float4 (128 cols = 32 float4s/row).
__global__ void init_out_bias(const float* __restrict__ bias,
                              float* __restrict__ out, int n4) {
  const float4* b4 = (const float4*)bias;
  float4*       o4 = (float4*)out;
  for (int i = blockIdx.x * blockDim.x + threadIdx.x; i < n4;
       i += gridDim.x * blockDim.x)
    o4[i] = b4[i & 31];
}

// One wave32 per edge (grid-stride over edges). Lane loads float4 of the
// source row (512B coalesced per edge), scales by edge_val, and scatter-adds
// with native global_atomic_add_f32. Random rows -> no LDS locality to mine;
// the 192MB L2 absorbs the RMW traffic since support+out fit entirely.
__global__ void spmm_scatter(const float* __restrict__ support,
                             const int* __restrict__ erow,
                             const int* __restrict__ ecol,
                             const float* __restrict__ eval,
                             float* __restrict__ out, int nEdges) {
  const int lane = threadIdx.x & 31;
  const int wpb  = blockDim.x >> 5;
  const int nW   = gridDim.x * wpb;
  for (int e = blockIdx.x * wpb + (threadIdx.x >> 5); e < nEdges; e += nW) {
    const int   r = erow[e];            // uniform per wave -> broadcast load
    const int   c = ecol[e];
    const float v = eval[e];
    const float4 m = ((const float4*)(support + (size_t)c * OUT_DIM))[lane];
    float* dst = out + (size_t)r * OUT_DIM + lane * 4;
    unsafeAtomicAdd(dst + 0, v * m.x);
    unsafeAtomicAdd(dst + 1, v * m.y);
    unsafeAtomicAdd(dst + 2, v * m.z);
    unsafeAtomicAdd(dst + 3, v * m.w);
  }
}

extern "C" void kernel_launch(void* const* d_in, const int* in_sizes, int n_in,
                              void* d_out, int out_size, void* d_ws, size_t ws_size,
                              hipStream_t stream) {
  const float* inp  = (const float*)d_in[0];   // [N, 128] fp32
  const int*   erow = (const int*)d_in[1];     // [E] i32
  const int*   ecol = (const int*)d_in[2];     // [E] i32
  const float* eval = (const float*)d_in[3];   // [E] fp32
  const float* W    = (const float*)d_in[4];   // [128, 128] fp32
  const float* bias = (const float*)d_in[5];   // [128] fp32

  float* out     = (float*)d_out;
  float* support = (float*)d_ws;               // [N, 128] fp32 scratch (51.2 MB)

  const int nNodes  = in_sizes[0] / IN_DIM;    // 100000 (multiple of 16)
  const int nEdges  = in_sizes[1];             // 3200000
  const int nStrips = nNodes / 16;             // 6250

  dim3 blk(256);                               // 8 wave32s per block

  gemm_support_wmma<<<(nStrips + 7) / 8, blk, 0, stream>>>(inp, W, support, nStrips);
  init_out_bias<<<2048, blk, 0, stream>>>(bias, out, (nNodes * OUT_DIM) / 4);
  spmm_scatter<<<4096, blk, 0, stream>>>(support, erow, ecol, eval, out, nEdges);
}